// CSA_36971078484033
// MI455X (gfx1250) — compile-verified
//
#include <hip/hip_runtime.h>

typedef __attribute__((ext_vector_type(2))) float v2f;
typedef __attribute__((ext_vector_type(8))) float v8f;

#define B_  32
#define C_  256
#define K_  64
#define K2  128          // concat(pos, neg) along K
#define HW  3136         // 56*56
#define HW4 784          // HW/4

// ---------------------------------------------------------------------------
// Kernel 1: label-conditional prototype mixing as a dense WMMA GEMM.
//   newMean = A_sel[32,128] x concat(Pmean_pos, Pmean_neg)[128,256]
//   newStd  = A_sel[32,128] x concat(Pstd_pos,  Pstd_neg )[128,256]
// One 16x16 output tile per wave; K accumulated 4 at a time with
// v_wmma_f32_16x16x4_f32. 64 waves total (8 blocks x 8 waves).
// ---------------------------------------------------------------------------
__global__ __launch_bounds__(256) void CSA_mix_wmma(
    const int*   __restrict__ label,   // [32]
    const float* __restrict__ w,       // [32,64]
    const float* __restrict__ Pmp,     // [64,256]
    const float* __restrict__ Psp,     // [64,256]
    const float* __restrict__ Pmn,     // [64,256]
    const float* __restrict__ Psn,     // [64,256]
    float*       __restrict__ newMean, // [32,256]
    float*       __restrict__ newStd)  // [32,256]
{
    __shared__ float As[B_][K2];       // 16 KB: label-selected A operand
    const int t = threadIdx.x;

    // Build A_sel (4096 elements, 16 per thread, no divergence).
    for (int i = t; i < B_ * K2; i += 256) {
        const int b = i >> 7;
        const int k = i & (K2 - 1);
        const bool pos = (label[b] == 0);
        const bool lowHalf = (k < K_);
        const float wv = w[b * K_ + (lowHalf ? k : k - K_)];
        As[b][k] = (lowHalf == pos) ? wv : 0.0f;
    }
    __syncthreads();

    const int wave = (blockIdx.x * 256 + t) >> 5;  // 0..63
    const int lane = t & 31;
    const int l    = lane & 15;                    // row/col within tile
    const int half = lane >> 4;                    // lane-half selects K pair / M+8

    const int mat   = wave >> 5;                   // 0 = mean, 1 = std
    const int tile  = wave & 31;
    const int mBase = (tile & 1) * 16;             // 2 M-tiles
    const int nBase = (tile >> 1) * 16;            // 16 N-tiles

    const float* Bpos = mat ? Psp : Pmp;
    const float* Bneg = mat ? Psn : Pmn;
    float*       D    = mat ? newStd : newMean;

    v8f acc = {};
    #pragma unroll 4
    for (int kk = 0; kk < K2 / 4; ++kk) {
        // A 16x4 layout: lanes 0-15 hold K={k0,k0+1}, lanes 16-31 K={k0+2,k0+3}
        const int k0 = kk * 4 + half * 2;
        v2f a, b;
        a.x = As[mBase + l][k0];
        a.y = As[mBase + l][k0 + 1];
        // B 4x16 layout mirrors A: VGPR r holds row K = k0+r (+2 for upper lanes)
        const float* r0 = (k0 < K_)     ? (Bpos + k0 * C_)           : (Bneg + (k0 - K_) * C_);
        const float* r1 = (k0 + 1 < K_) ? (Bpos + (k0 + 1) * C_)     : (Bneg + (k0 + 1 - K_) * C_);
        b.x = r0[nBase + l];
        b.y = r1[nBase + l];
        acc = __builtin_amdgcn_wmma_f32_16x16x4_f32(
            /*neg_a=*/false, a, /*neg_b=*/false, b,
            /*c_mod=*/(short)0, acc, /*reuse_a=*/false, /*reuse_b=*/false);
    }

    // C/D layout: VGPR r -> M = mBase + r + 8*half, N = nBase + l
    #pragma unroll
    for (int r = 0; r < 8; ++r)
        D[(mBase + r + 8 * half) * C_ + nBase + l] = acc[r];
}

// ---------------------------------------------------------------------------
// Kernel 2: fused instance-norm + affine. One block per (b,c) plane.
// Single read of x, single write of out -> ~206 MB total @ 23.3 TB/s.
// ---------------------------------------------------------------------------
__global__ __launch_bounds__(256) void CSA_inorm(
    const float* __restrict__ x,
    const float* __restrict__ newMean,
    const float* __restrict__ newStd,
    float*       __restrict__ out)
{
    const int plane = blockIdx.x;                          // b*C + c
    const float4* xp = (const float4*)(x + (size_t)plane * HW);
    float4*       op = (float4*)(out + (size_t)plane * HW);
    const int t = threadIdx.x;

    // 784 float4 per plane: 3 full rounds of 256 + 16-wide tail.
    float4 v0 = xp[t];
    float4 v1 = xp[t + 256];
    float4 v2 = xp[t + 512];
    float4 v3 = (t < 16) ? xp[t + 768] : make_float4(0.f, 0.f, 0.f, 0.f);

    float s  = (v0.x + v0.y + v0.z + v0.w) + (v1.x + v1.y + v1.z + v1.w)
             + (v2.x + v2.y + v2.z + v2.w) + (v3.x + v3.y + v3.z + v3.w);
    float sq = (v0.x*v0.x + v0.y*v0.y + v0.z*v0.z + v0.w*v0.w)
             + (v1.x*v1.x + v1.y*v1.y + v1.z*v1.z + v1.w*v1.w)
             + (v2.x*v2.x + v2.y*v2.y + v2.z*v2.z + v2.w*v2.w)
             + (v3.x*v3.x + v3.y*v3.y + v3.z*v3.z + v3.w*v3.w);

    // wave32 reduction
    #pragma unroll
    for (int m = 16; m >= 1; m >>= 1) {
        s  += __shfl_xor(s,  m, 32);
        sq += __shfl_xor(sq, m, 32);
    }
    __shared__ float ws_s[8], ws_q[8];
    if ((t & 31) == 0) { ws_s[t >> 5] = s; ws_q[t >> 5] = sq; }
    __syncthreads();
    float tot = 0.f, totq = 0.f;
    #pragma unroll
    for (int i = 0; i < 8; ++i) { tot += ws_s[i]; totq += ws_q[i]; }

    const float mean = tot * (1.0f / HW);
    const float var  = (totq - (float)HW * mean * mean) * (1.0f / (HW - 1)); // ddof=1
    const float rstd = rsqrtf(var + 1e-5f);
    const float a = newStd[plane] * rstd;          // scale
    const float c = newMean[plane] - mean * a;     // shift

    v0.x = fmaf(v0.x, a, c); v0.y = fmaf(v0.y, a, c); v0.z = fmaf(v0.z, a, c); v0.w = fmaf(v0.w, a, c);
    v1.x = fmaf(v1.x, a, c); v1.y = fmaf(v1.y, a, c); v1.z = fmaf(v1.z, a, c); v1.w = fmaf(v1.w, a, c);
    v2.x = fmaf(v2.x, a, c); v2.y = fmaf(v2.y, a, c); v2.z = fmaf(v2.z, a, c); v2.w = fmaf(v2.w, a, c);
    op[t]       = v0;
    op[t + 256] = v1;
    op[t + 512] = v2;
    if (t < 16) {
        v3.x = fmaf(v3.x, a, c); v3.y = fmaf(v3.y, a, c);
        v3.z = fmaf(v3.z, a, c); v3.w = fmaf(v3.w, a, c);
        op[t + 768] = v3;
    }
}

extern "C" void kernel_launch(void* const* d_in, const int* in_sizes, int n_in,
                              void* d_out, int out_size, void* d_ws, size_t ws_size,
                              hipStream_t stream) {
    const float* x     = (const float*)d_in[0];
    const int*   label = (const int*)  d_in[1];
    const float* w     = (const float*)d_in[2];
    const float* Pmp   = (const float*)d_in[3];
    const float* Psp   = (const float*)d_in[4];
    const float* Pmn   = (const float*)d_in[5];
    const float* Psn   = (const float*)d_in[6];
    float* out = (float*)d_out;

    float* newMean = (float*)d_ws;            // [32,256] = 32 KB
    float* newStd  = newMean + B_ * C_;       // [32,256] = 32 KB

    CSA_mix_wmma<<<8, 256, 0, stream>>>(label, w, Pmp, Psp, Pmn, Psn, newMean, newStd);
    CSA_inorm<<<B_ * C_, 256, 0, stream>>>(x, newMean, newStd, out);
}